// RegularRenderer_34497177322118
// MI455X (gfx1250) — compile-verified
//
#include <hip/hip_runtime.h>
#include <hip/hip_bf16.h>
#include <math.h>

// NeRF regular-grid volume renderer, MI455X (gfx1250, wave32).
// Bandwidth-bound: ~202 MB mandatory traffic -> ~8.7 us at 23.3 TB/s.
// This version streams tiles through LDS with gfx1250 async copies
// (GLOBAL_LOAD_ASYNC_TO_LDS_B128, ASYNCcnt-tracked) and double-buffers so
// every wave keeps a full next tile in flight while computing the current
// one -> high memory-level parallelism independent of occupancy.

#ifndef EPS_T
#define EPS_T 1e-10f
#endif

typedef __attribute__((ext_vector_type(2))) float v2f;

static constexpr int kN            = 192;   // samples per ray
static constexpr int kPerLane      = 6;     // 192 / 32 lanes
static constexpr int kThreads      = 256;   // 8 waves
static constexpr int kRaysPerChunk = 16;    // 2 rays per wave
static constexpr int kChunksPerBlk = 8;

// Per-chunk byte sizes (float32)
static constexpr int kDensBytes = kRaysPerChunk * kN * 4;   // 12288
static constexpr int kRgbBytes  = kRaysPerChunk * kN * 12;  // 36864
static constexpr int kDensB128PerThread = kDensBytes / 16 / kThreads; // 3
static constexpr int kRgbB128PerThread  = kRgbBytes  / 16 / kThreads; // 9
// 12 async ops in flight per thread per chunk
static constexpr int kLdsFloats = (2 * kDensBytes + 2 * kRgbBytes) / 4; // 24576 (96 KB)

// ---------------------------------------------------------------------------
// gfx1250 async global->LDS copy. Single __shared__ object => its LDS offset
// is 0, so byte offsets into it are valid LDS addresses (VDST operand).
// LDS address is the VDST value plus the instruction offset (ISA 15.18.3, ASYNCcnt-tracked)
// ---------------------------------------------------------------------------
__device__ __forceinline__ void async_b128(uint32_t lds_byte_off, const void* gaddr) {
    asm volatile("global_load_async_to_lds_b128 %0, %1, off"
                 :
                 : "v"(lds_byte_off), "v"(gaddr)
                 : "memory");
}

__device__ __forceinline__ void wait_async_le12() {
#if __has_builtin(__builtin_amdgcn_s_wait_asynccnt)
    __builtin_amdgcn_s_wait_asynccnt(12);
#else
    asm volatile("s_wait_asynccnt 0xc" ::: "memory");
#endif
}
__device__ __forceinline__ void wait_async_le0() {
#if __has_builtin(__builtin_amdgcn_s_wait_asynccnt)
    __builtin_amdgcn_s_wait_asynccnt(0);
#else
    asm volatile("s_wait_asynccnt 0x0" ::: "memory");
#endif
}

// Issue one chunk's copies: 3 + 9 = 12 b128 per thread, fully coalesced
// (512 B per wave per instruction), no predication (exact tiling).
__device__ __forceinline__ void issue_chunk(
    const char* dens_g, const char* rgb_g,   // chunk-based global pointers
    int buf, int tid)
{
    const uint32_t dens_lds = (uint32_t)buf * kDensBytes;                   // 0 / 12288
    const uint32_t rgb_lds  = 2 * kDensBytes + (uint32_t)buf * kRgbBytes;   // 24576 / 61440
#pragma unroll
    for (int k = 0; k < kDensB128PerThread; ++k) {
        const uint32_t off = (uint32_t)(tid + kThreads * k) * 16u;
        async_b128(dens_lds + off, dens_g + off);
    }
#pragma unroll
    for (int k = 0; k < kRgbB128PerThread; ++k) {
        const uint32_t off = (uint32_t)(tid + kThreads * k) * 16u;
        async_b128(rgb_lds + off, rgb_g + off);
    }
}

// Per-ray math: exclusive cumprod of t = e^{-d*delta}+eps
// via per-lane product + wave32 scan, then fused weight*rgb reduction.
__device__ __forceinline__ void render_ray_from_lds(
    const float* __restrict__ dens,   // this ray's 192 densities (LDS)
    const float* __restrict__ rgb,    // this ray's 192*3 colors   (LDS)
    float delta, int lane, int ray, float* __restrict__ out)
{
    const v2f* dp = reinterpret_cast<const v2f*>(dens + lane * kPerLane);
    v2f d01 = dp[0], d23 = dp[1], d45 = dp[2];
    float d[kPerLane] = {d01.x, d01.y, d23.x, d23.y, d45.x, d45.y};

    float e[kPerLane], t[kPerLane];
    float q = 1.0f;
#pragma unroll
    for (int j = 0; j < kPerLane; ++j) {
        e[j] = __expf(-d[j] * delta);
        t[j] = e[j] + EPS_T;
        q *= t[j];
    }

    float v = q;
#pragma unroll
    for (int off = 1; off < 32; off <<= 1) {
        float o = __shfl_up(v, off, 32);
        if (lane >= off) v *= o;
    }
    float carry = __shfl_up(v, 1, 32);
    if (lane == 0) carry = 1.0f;

    const v2f* rp = reinterpret_cast<const v2f*>(rgb + lane * kPerLane * 3);
    v2f rv[9];
#pragma unroll
    for (int k = 0; k < 9; ++k) rv[k] = rp[k];
    const float* c = reinterpret_cast<const float*>(rv);

    float s0 = 0.0f, s1 = 0.0f, s2 = 0.0f;
    float pre = carry;
#pragma unroll
    for (int j = 0; j < kPerLane; ++j) {
        const float alpha = 1.0f - e[j];
        const float w = pre * alpha;
        s0 = fmaf(w, c[3 * j + 0], s0);
        s1 = fmaf(w, c[3 * j + 1], s1);
        s2 = fmaf(w, c[3 * j + 2], s2);
        pre *= t[j];
    }
#pragma unroll
    for (int off = 16; off > 0; off >>= 1) {
        s0 += __shfl_down(s0, off, 32);
        s1 += __shfl_down(s1, off, 32);
        s2 += __shfl_down(s2, off, 32);
    }
    if (lane == 0) {
        out[(size_t)ray * 3 + 0] = s0;
        out[(size_t)ray * 3 + 1] = s1;
        out[(size_t)ray * 3 + 2] = s2;
    }
}

__global__ __launch_bounds__(kThreads) void RegularRenderer_async_kernel(
    const float* __restrict__ rgbs,
    const float* __restrict__ densities,
    const float* __restrict__ points,
    float* __restrict__ out,
    int nchunks)
{
    __shared__ __align__(16) float lds_f[kLdsFloats];  // single LDS object @ offset 0

    const int tid  = threadIdx.x;
    const int lane = tid & 31;
    const int wave = tid >> 5;

    const int c0 = blockIdx.x * kChunksPerBlk;
    if (c0 >= nchunks) return;  // uniform per block
    const int cend = min(c0 + kChunksPerBlk, nchunks);

    // delta = ||pts[0,1,:] - pts[0,0,:]||, 6 broadcast loads (L2-resident)
    const float dx = points[3] - points[0];
    const float dy = points[4] - points[1];
    const float dz = points[5] - points[2];
    const float delta = sqrtf(dx * dx + dy * dy + dz * dz);

    const char* dens_base = (const char*)densities;
    const char* rgb_base  = (const char*)rgbs;

    // Prologue: fill buffer 0
    issue_chunk(dens_base + (size_t)c0 * kDensBytes,
                rgb_base  + (size_t)c0 * kRgbBytes, 0, tid);

    for (int c = c0; c < cend; ++c) {
        const int cur = (c - c0) & 1;
        if (c + 1 < cend) {
            issue_chunk(dens_base + (size_t)(c + 1) * kDensBytes,
                        rgb_base  + (size_t)(c + 1) * kRgbBytes, cur ^ 1, tid);
            wait_async_le12();   // in-order => current chunk's 12 ops are done
        } else {
            wait_async_le0();
        }
        __syncthreads();         // join all waves' copies before reading LDS

        const float* dens_lds = lds_f + cur * (kDensBytes / 4);
        const float* rgb_lds  = lds_f + 2 * (kDensBytes / 4) + cur * (kRgbBytes / 4);
#pragma unroll
        for (int s = 0; s < 2; ++s) {
            const int rl  = wave * 2 + s;            // 0..15 within chunk
            const int ray = c * kRaysPerChunk + rl;
            render_ray_from_lds(dens_lds + rl * kN,
                                rgb_lds  + rl * kN * 3,
                                delta, lane, ray, out);
        }
        __syncthreads();         // protect buffer before next refill
    }
}

// Remainder rays (R % 16 != 0): direct-load fallback, one wave per ray.
__global__ __launch_bounds__(kThreads) void RegularRenderer_tail_kernel(
    const float* __restrict__ rgbs,
    const float* __restrict__ densities,
    const float* __restrict__ points,
    float* __restrict__ out,
    int ray0, int R)
{
    const int lane = threadIdx.x & 31;
    const int wave = threadIdx.x >> 5;
    const int ray  = ray0 + blockIdx.x * 8 + wave;
    if (ray >= R) return;

    const float dx = points[3] - points[0];
    const float dy = points[4] - points[1];
    const float dz = points[5] - points[2];
    const float delta = sqrtf(dx * dx + dy * dy + dz * dz);

    const size_t nbase = (size_t)ray * kN + (size_t)lane * kPerLane;
    render_ray_from_lds(densities + (size_t)ray * kN - 0 + 0 + 0 * nbase + 0 + 0 + 0
                        , rgbs + (size_t)ray * kN * 3, delta, lane, ray, out);
}

extern "C" void kernel_launch(void* const* d_in, const int* in_sizes, int n_in,
                              void* d_out, int out_size, void* d_ws, size_t ws_size,
                              hipStream_t stream) {
    (void)n_in; (void)out_size; (void)d_ws; (void)ws_size;
    const float* rgbs      = (const float*)d_in[0];
    const float* densities = (const float*)d_in[1];
    const float* points    = (const float*)d_in[2];
    float*       out       = (float*)d_out;

    const int R       = in_sizes[1] / kN;        // densities is (R, N)
    const int nchunks = R / kRaysPerChunk;       // full 16-ray chunks
    const int blocks  = (nchunks + kChunksPerBlk - 1) / kChunksPerBlk;

    if (blocks > 0) {
        RegularRenderer_async_kernel<<<blocks, kThreads, 0, stream>>>(
            rgbs, densities, points, out, nchunks);
    }
    const int done = nchunks * kRaysPerChunk;
    if (done < R) {  // never taken for R = 65536
        const int rem = R - done;
        RegularRenderer_tail_kernel<<<(rem + 7) / 8, kThreads, 0, stream>>>(
            rgbs, densities, points, out, done, R);
    }
}